// Raster_87205015978273
// MI455X (gfx1250) — compile-verified
//
#include <hip/hip_runtime.h>
#include <hip/hip_bf16.h>
#include <math.h>

typedef __attribute__((ext_vector_type(2))) float v2f;
typedef __attribute__((ext_vector_type(4))) float v4f;
typedef __attribute__((ext_vector_type(8))) float v8f;

#define PATCH 10

// Per-wave LDS: cdf[3][12] (36) + qw0[12] + tile10[100] + pad = 160 floats (16B-aligned stride)
#define WAVE_LDS 160

__global__ __launch_bounds__(256) void Raster_87205015978273_kernel(
    const float* __restrict__ sigma,   // [N,3]
    const float* __restrict__ time,    // [N]
    const float* __restrict__ charge,  // [N]
    const float* __restrict__ tail,    // [N,3]
    const float* __restrict__ gs,      // [3]
    const float* __restrict__ nsig,    // [1]
    float* __restrict__ out,           // rasters [N,1000] then offsets [N,3]
    int nD)
{
    __shared__ float smem[8 * WAVE_LDS];

    const int tid  = blockIdx.x * blockDim.x + threadIdx.x;
    const int n    = tid >> 5;          // wave id == depo id
    const int lane = threadIdx.x & 31;  // wave32
    if (n >= nD) return;

    float* W       = smem + ((threadIdx.x >> 5) * WAVE_LDS);
    float* Wcdf    = W;        // 3 * 12
    float* Wqw0    = W + 36;   // 12
    float* Wtile10 = W + 48;   // 100, base offset 192B -> 16B aligned

    // ---- scalar per-depo parameters (broadcast loads, hit cache) ----
    const float c0 = tail[n * 3 + 1];   // _transform: [tail.y, tail.z, time]
    const float c1 = tail[n * 3 + 2];
    const float c2 = time[n];
    const float s0 = sigma[n * 3 + 0];
    const float s1 = sigma[n * 3 + 1];
    const float s2 = sigma[n * 3 + 2];
    const float h0 = gs[0], h1 = gs[1], h2 = gs[2];
    const float ns = nsig[0];
    const float q  = charge[n];

    const float im0 = floorf((c0 - ns * s0) / h0);
    const float im1 = floorf((c1 - ns * s1) / h1);
    const float im2 = floorf((c2 - ns * s2) / h2);

    // ---- 33 CDF edge values spread over all 32 lanes (1 erff each + 1 extra) ----
    {
        const int d = (lane < 11) ? 0 : (lane < 22) ? 1 : 2;
        const int e = lane - d * 11;
        const float imd = (d == 0) ? im0 : (d == 1) ? im1 : im2;
        const float hd  = (d == 0) ? h0  : (d == 1) ? h1  : h2;
        const float cd  = (d == 0) ? c0  : (d == 1) ? c1  : c2;
        const float sd  = (d == 0) ? s0  : (d == 1) ? s1  : s2;
        const float invd = 0.70710678118f / sd;    // 1/(sqrt(2)*sigma)
        Wcdf[d * 12 + e] =
            0.5f * (1.0f + erff(((imd + (float)e) * hd - cd) * invd));
        if (lane == 0) {   // 33rd value: (d=2, e=10)
            const float inv2 = 0.70710678118f / s2;
            Wcdf[2 * 12 + 10] =
                0.5f * (1.0f + erff(((im2 + 10.0f) * h2 - c2) * inv2));
        }
    }
    asm volatile("s_wait_dscnt 0" ::: "memory");  // same-wave LDS ordering

    // ---- per-dim weights; WMMA operands for outer(w1, w2) ----
    const int half = lane >> 4;
    const int col  = lane & 15;
    float aval = 0.0f;
    float bval = 0.0f;
    if (lane < PATCH) {   // lanes 0..9 -> A rows M=0..9, K=0 (lanes>=16 hold K=2,3: zero)
        aval = Wcdf[12 + lane + 1] - Wcdf[12 + lane];           // w1[j]
        Wqw0[lane] = q * (Wcdf[lane + 1] - Wcdf[lane]);         // q * w0[i]
    }
    if (col < PATCH) {
        bval = Wcdf[24 + col + 1] - Wcdf[24 + col];             // w2[k]
    }

    // A: only K=0 nonzero (w1 in lanes 0..15 / VGPR0). B: w2 replicated into every
    // K slot -- only the K=0 row participates since A's other columns are zero.
    v2f a; a.x = aval; a.y = 0.0f;
    v2f b; b.x = bval; b.y = bval;
    v8f acc = {};
    // D(j,k) = w1[j] * w2[k]   (16x16 tile, EXEC all-ones at this point)
    acc = __builtin_amdgcn_wmma_f32_16x16x4_f32(
        /*neg_a=*/false, a, /*neg_b=*/false, b,
        /*c_mod=*/(short)0, acc, /*reuse_a=*/false, /*reuse_b=*/false);

    // ---- compact D tile into stride-10 layout: tile10[j*10+k], j,k < 10 ----
    // ISA C/D layout: VGPR r, lane l holds elem (j = r + 8*(l>=16), k = l%16).
    // Rows j >= 10 are zero AND out of tile10 bounds -> must be predicated off.
    {
        const int base = half * 80 + col;   // (8*half)*10 + col
#pragma unroll
        for (int r = 0; r < 8; ++r) {
            const bool ok = (col < PATCH) && (half == 0 || r < 2);  // j < 10
            if (ok) Wtile10[base + r * 10] = acc[r];
        }
    }
    asm volatile("s_wait_dscnt 0" ::: "memory");

    // ---- stream 1000 outputs as 8 wave-wide b128 NT stores (512B each) ----
    // 400MB output stream > 192MB L2 -> non-temporal. i = f/100 is uniform within
    // each lane's 4-element group (100 % 4 == 0); r..r+3 contiguous in tile10.
    float* rast = out + (size_t)n * 1000;
#pragma unroll
    for (int t = 0; t < 8; ++t) {
        const int f = t * 128 + 4 * lane;     // flat output index, 16B aligned
        if (f < 1000) {                        // t=7: lanes 0..25 active
            const int i = f / 100;
            const int r = f - i * 100;
            const v4f tv = *(const v4f*)(Wtile10 + r);   // ds_load_b128
            const float qv = Wqw0[i];                     // broadcast ds_load
            v4f o;
            o.x = qv * tv.x; o.y = qv * tv.y;
            o.z = qv * tv.z; o.w = qv * tv.w;
            __builtin_nontemporal_store(o, (v4f*)(rast + f));  // global_store_b128 NT
        }
    }

    // ---- offsets (3 ints per depo, emitted into the float output region) ----
    if (lane < 3) {
        const float im = (lane == 0) ? im0 : (lane == 1) ? im1 : im2;
        out[(size_t)nD * 1000 + (size_t)n * 3 + lane] = im;
    }
}

extern "C" void kernel_launch(void* const* d_in, const int* in_sizes, int n_in,
                              void* d_out, int out_size, void* d_ws, size_t ws_size,
                              hipStream_t stream) {
    const float* sigma  = (const float*)d_in[0];
    const float* time_  = (const float*)d_in[1];
    const float* charge = (const float*)d_in[2];
    const float* tail   = (const float*)d_in[3];
    const float* gs     = (const float*)d_in[4];
    const float* nsig   = (const float*)d_in[5];
    float* out = (float*)d_out;

    const int nD = in_sizes[1];               // N (time is [N])
    const int waves_per_block = 256 / 32;     // 8 depos per block
    const int blocks = (nD + waves_per_block - 1) / waves_per_block;
    Raster_87205015978273_kernel<<<blocks, 256, 0, stream>>>(
        sigma, time_, charge, tail, gs, nsig, out, nD);
}